// LossFunction_8400956031154
// MI455X (gfx1250) — compile-verified
//
#include <hip/hip_runtime.h>
#include <stdint.h>

// SSD MultiBox loss for MI455X (gfx1250).
//
// Roofline: ~140MB of input @ 23.3TB/s ≈ 6us floor; ~100 MFLOP of f32 VALU is
// negligible -> pure bandwidth problem, WMMA not applicable (no matmul shape).
// CDNA5 paths used: Tensor Data Mover (tensor_load_to_lds, double-buffered,
// retired with s_wait_tensorcnt) to stage the 94MB class-logit stream into
// LDS; DS atomics for block reductions; exact radix-select for hard-negative
// mining on an LDS-resident CE row (35KB of the 320KB WGP LDS).
//
// Fast path (workspace permitting): phase A = 1152 blocks compute CE+SmoothL1
// with overlapped TDM DMA, CE -> 4.5MB ws (L2-resident, 192MB L2); phase B =
// 128 blocks mine hard negatives per row; phase C = mean. Fallback: one
// monolithic 128-block kernel (512B ws).

constexpr int kB = 128;
constexpr int kA = 8732;
constexpr int kC = 21;
constexpr int NT = 256;                   // threads per block (8 waves, wave32)
constexpr int CH = 256;                   // anchors per TDM chunk
constexpr int SEG = 1024;                 // anchors per phase-A block
constexpr int NSEG = (kA + SEG - 1) / SEG;        // 9
constexpr int NMASKW = (kA + 31) / 32;            // 273 mask words

#if defined(__has_builtin)
#if __has_builtin(__builtin_amdgcn_tensor_load_to_lds) && \
    __has_builtin(__builtin_amdgcn_s_wait_tensorcnt)
#define USE_TDM 1
#endif
#endif

#ifdef USE_TDM
typedef unsigned int v4u __attribute__((ext_vector_type(4)));
typedef int          v8i __attribute__((ext_vector_type(8)));
typedef int          v4i __attribute__((ext_vector_type(4)));

// 1D tile load: nelem 32-bit words, global -> LDS, via TDM descriptor (D#).
// Field packing per CDNA5 ISA ch.8 (group0 128b, group1 256b); 6-arg builtin.
__device__ __forceinline__ void tdm_load_1d(uint32_t lds_off, const void* gptr,
                                            uint32_t nelem) {
  const uint64_t ga = (uint64_t)(uintptr_t)gptr;
  v4u g0;
  g0[0] = 1u;                                   // count=1, user mode
  g0[1] = lds_off;                              // lds_addr[31:0]
  g0[2] = (uint32_t)(ga & 0xFFFFFFFFu);         // global_addr[31:0]
  g0[3] = (uint32_t)((ga >> 32) & 0x01FFFFFFu)  // global_addr[56:32]
        | (2u << 30);                           // type=2 ("image")
  v8i g1;
  g1[0] = (int)(2u << 16);                      // data_size=2 (4 bytes)
  g1[1] = (int)((nelem & 0xFFFFu) << 16);       // tensor_dim0[15:0] @ [31:16]
  g1[2] = (int)(((nelem >> 16) & 0xFFFFu)       // tensor_dim0[31:16]
        | (1u << 16));                          // tensor_dim1[15:0] = 1
  g1[3] = (int)((nelem & 0xFFFFu) << 16);       // tile_dim0 = nelem (<=5376)
  g1[4] = (int)1;                               // tile_dim1 = 1, tile_dim2 = 0
  g1[5] = (int)nelem;                           // tensor_dim0_stride[31:0]
  g1[6] = (int)((nelem & 0xFFFFu) << 16);       // stride0[47:32]=0, stride1[15:0]
  g1[7] = (int)((nelem >> 16) & 0xFFFFu);       // stride1[47:16]
  v4i g2 = {0, 0, 0, 0};
  v4i g3 = {0, 0, 0, 0};
  v8i g4 = {0, 0, 0, 0, 0, 0, 0, 0};
  __builtin_amdgcn_tensor_load_to_lds(g0, g1, g2, g3, g4, 0);
}
#endif

// ---------------- shared per-anchor math ----------------

__device__ __forceinline__ float cross_entropy_from_lds(const float* xr,
                                                        int lbl) {
  float m = xr[0];
#pragma unroll
  for (int c = 1; c < kC; ++c) m = fmaxf(m, xr[c]);
  float s = 0.f;
#pragma unroll
  for (int c = 0; c < kC; ++c) s += __expf(xr[c] - m);
  const int li = min(max(lbl, 0), kC - 1);
  return __logf(s) + m - xr[li];  // -log_softmax gathered at label
}

__device__ __forceinline__ float smooth_l1_4(const float4& po, const float4& pl) {
  float sl = 0.f;
  { float d = po.x - pl.x, ad = fabsf(d); sl += (ad < 1.f) ? 0.5f*d*d : ad - 0.5f; }
  { float d = po.y - pl.y, ad = fabsf(d); sl += (ad < 1.f) ? 0.5f*d*d : ad - 0.5f; }
  { float d = po.z - pl.z, ad = fabsf(d); sl += (ad < 1.f) ? 0.5f*d*d : ad - 0.5f; }
  { float d = po.w - pl.w, ad = fabsf(d); sl += (ad < 1.f) ? 0.5f*d*d : ad - 0.5f; }
  return sl;
}

// con_neg bit pattern (CE >= 0 so uint order == float order; clamp -eps)
__device__ __forceinline__ uint32_t conneg_bits(const float* s_con,
                                                const uint32_t* s_maskw, int a) {
  const bool pm = (s_maskw[a >> 5] >> (a & 31)) & 1u;
  float v = pm ? 0.f : s_con[a];
  v = fmaxf(v, 0.f);
  return __float_as_uint(v);
}

// ---------------- hard negative mining (block-cooperative) ----------------
// Exact reproduction of: rank = argsort(argsort(-con_neg)); neg_mask = rank<k;
// returns sum(con[neg_mask]). Uniform branches; barriers inside.
__device__ float hard_negative_sum(const float* s_con, const uint32_t* s_maskw,
                                   uint32_t* s_hist, int* s_scan,
                                   float* sredf, int* sredi,
                                   uint32_t* s_prefix, int* s_rem,
                                   int k, int nz, float neg_tot, float pos_con,
                                   int tid) {
  if (k >= kA) {
    // every anchor selected by neg_mask -> sum of all con
    return neg_tot + pos_con;
  }
  if (k <= nz) {
    // radix-select threshold among strictly-positive con_neg values
    uint32_t prefix = 0;
    int remaining = k;
    for (int pass = 3; pass >= 0; --pass) {
      const int shift = pass * 8;
      for (int i = tid; i < 256; i += NT) s_hist[i] = 0u;
      __syncthreads();
      for (int a = tid; a < kA; a += NT) {
        const uint32_t u = conneg_bits(s_con, s_maskw, a);
        const bool cand =
            (pass == 3) || ((u >> (shift + 8)) == (prefix >> (shift + 8)));
        if (cand) atomicAdd(&s_hist[(u >> shift) & 255u], 1u);
      }
      __syncthreads();
      if (tid == 0) {
        int cum = 0; uint32_t sel = 0;
        for (int bin = 255; bin >= 0; --bin) {
          const int h = (int)s_hist[bin];
          if (cum + h >= remaining) { sel = (uint32_t)bin; break; }
          cum += h;
        }
        *s_prefix = prefix | (sel << shift);
        *s_rem    = remaining - cum;
      }
      __syncthreads();
      prefix = *s_prefix;
      remaining = *s_rem;
      __syncthreads();
    }
    const uint32_t T = prefix;
    if (tid == 0) { sredf[3] = 0.f; sredi[2] = 0; }
    __syncthreads();
    float sgt = 0.f; int cgt = 0;
    for (int a = tid; a < kA; a += NT) {
      const uint32_t u = conneg_bits(s_con, s_maskw, a);
      if (u > T) { sgt += __uint_as_float(u); ++cgt; }
    }
    atomicAdd(&sredf[3], sgt);
    atomicAdd(&sredi[2], cgt);
    __syncthreads();
    return sredf[3] + (float)(k - sredi[2]) * __uint_as_float(T);
  }
  // nz < k < A: all nonzero con_neg selected, plus first (k-nz) zero-valued
  // entries in index order (stable argsort semantics); add their full con.
  const int m = k - nz;
  if (tid == 0) sredf[3] = 0.f;
  __syncthreads();
  float zadd = 0.f;
  int offset = 0;
  for (int base = 0; base < kA; base += NT) {
    const int a = base + tid;
    int isz = 0; float cv = 0.f;
    if (a < kA) {
      if (conneg_bits(s_con, s_maskw, a) == 0u) { isz = 1; cv = s_con[a]; }
    }
    s_scan[tid] = isz;
    __syncthreads();
    for (int d = 1; d < NT; d <<= 1) {  // Hillis-Steele inclusive scan
      const int mine = s_scan[tid];
      const int left = (tid >= d) ? s_scan[tid - d] : 0;
      __syncthreads();
      s_scan[tid] = mine + left;
      __syncthreads();
    }
    const int incl = s_scan[tid];
    const int tot  = s_scan[NT - 1];
    const int excl = incl - isz;
    if (isz && (offset + excl) < m) zadd += cv;
    __syncthreads();
    offset += tot;
  }
  atomicAdd(&sredf[3], zadd);
  __syncthreads();
  return neg_tot + sredf[3];
}

// =================== fast path: phase A (CE + SmoothL1) ===================
// grid (NSEG, kB); double-buffered TDM staging; writes con[] + row atomics.
__global__ __launch_bounds__(NT)
void ssd_phaseA(const float* __restrict__ loc_out,
                const float* __restrict__ cls_out,
                const float* __restrict__ loc_lab,
                const int*   __restrict__ cls_lab,
                float* __restrict__ ws_con,      // [kB*kA]
                float* __restrict__ accf,        // [3*kB]: loc, posc, negc
                int*   __restrict__ acci) {      // [2*kB]: pos, nz
  __shared__ float s_logits[2][CH * kC];         // 2 x 21504 B
  __shared__ float sredf[3];
  __shared__ int   sredi[2];

  const int seg = blockIdx.x;
  const int b   = blockIdx.y;
  const int tid = threadIdx.x;
  const size_t rowbase  = (size_t)b * kA;
  const int    segstart = seg * SEG;
  const int    segN     = min(SEG, kA - segstart);
  const int    nch      = (segN + CH - 1) / CH;

  if (tid == 0) { sredf[0] = sredf[1] = sredf[2] = 0.f; sredi[0] = sredi[1] = 0; }

  float acc_loc = 0.f, acc_posc = 0.f, acc_negc = 0.f;
  int   acc_pos = 0,   acc_nz = 0;

#ifdef USE_TDM
  if (tid < 32) {  // wave 0 issues DMAs; tensor ops are EXEC-independent
    const int n0 = min(CH, segN) * kC;
    tdm_load_1d((uint32_t)(uintptr_t)(&s_logits[0][0]),
                cls_out + (rowbase + segstart) * (size_t)kC, (uint32_t)n0);
  }
#endif

  for (int c = 0; c < nch; ++c) {
    const int base = segstart + c * CH;
    const int cn   = min(CH, segstart + segN - base);
#ifdef USE_TDM
    if (tid < 32) {
      if (c + 1 < nch) {  // prefetch next chunk into the other buffer
        const int nb = segstart + (c + 1) * CH;
        const int nn = min(CH, segstart + segN - nb) * kC;
        tdm_load_1d((uint32_t)(uintptr_t)(&s_logits[(c + 1) & 1][0]),
                    cls_out + (rowbase + nb) * (size_t)kC, (uint32_t)nn);
        __builtin_amdgcn_s_wait_tensorcnt(1);  // retire chunk c, keep c+1 flying
      } else {
        __builtin_amdgcn_s_wait_tensorcnt(0);
      }
    }
#else
    for (int i = tid; i < cn * kC; i += NT)
      s_logits[c & 1][i] = cls_out[(rowbase + base) * (size_t)kC + i];
#endif
    __syncthreads();

    if (tid < cn) {
      const int a   = base + tid;
      const int lbl = cls_lab[rowbase + a];
      const float ce =
          cross_entropy_from_lds(&s_logits[c & 1][tid * kC], lbl);
      ws_con[rowbase + a] = ce;
      if (lbl > 0) {
        acc_pos  += 1;
        acc_posc += ce;
        acc_loc  += smooth_l1_4(
            reinterpret_cast<const float4*>(loc_out)[rowbase + a],
            reinterpret_cast<const float4*>(loc_lab)[rowbase + a]);
      } else {
        acc_negc += ce;
        if (ce > 0.f) acc_nz += 1;
      }
    }
    __syncthreads();  // buffer c&1 free before it is re-issued next iteration
  }

  atomicAdd(&sredf[0], acc_loc);
  atomicAdd(&sredf[1], acc_posc);
  atomicAdd(&sredf[2], acc_negc);
  atomicAdd(&sredi[0], acc_pos);
  atomicAdd(&sredi[1], acc_nz);
  __syncthreads();
  if (tid == 0) {
    atomicAdd(&accf[0 * kB + b], sredf[0]);
    atomicAdd(&accf[1 * kB + b], sredf[1]);
    atomicAdd(&accf[2 * kB + b], sredf[2]);
    atomicAdd(&acci[0 * kB + b], sredi[0]);
    atomicAdd(&acci[1 * kB + b], sredi[1]);
  }
}

// =================== fast path: phase B (mining per row) ===================
__global__ __launch_bounds__(NT)
void ssd_phaseB(const float* __restrict__ ws_con,
                const int*   __restrict__ cls_lab,
                const float* __restrict__ accf,
                const int*   __restrict__ acci,
                float* __restrict__ rows) {
  __shared__ float    s_con[kA];
  __shared__ uint32_t s_maskw[NMASKW];
  __shared__ uint32_t s_hist[256];
  __shared__ int      s_scan[NT];
  __shared__ float    sredf[4];
  __shared__ int      sredi[3];
  __shared__ uint32_t s_prefix;
  __shared__ int      s_rem;

  const int b   = blockIdx.x;
  const int tid = threadIdx.x;
  const size_t rowbase = (size_t)b * kA;

  for (int i = tid; i < NMASKW; i += NT) s_maskw[i] = 0u;
  __syncthreads();
  for (int a = tid; a < kA; a += NT) {
    s_con[a] = ws_con[rowbase + a];            // L2-hot (4.5MB total)
    if (cls_lab[rowbase + a] > 0) atomicOr(&s_maskw[a >> 5], 1u << (a & 31));
  }
  __syncthreads();

  const float loc_loss = accf[0 * kB + b];
  const float pos_con  = accf[1 * kB + b];
  const float neg_tot  = accf[2 * kB + b];
  const int   pos      = acci[0 * kB + b];
  const int   nz       = acci[1 * kB + b];

  int k = min(3 * pos, kA);
  if (k == 0) k = 3;
  const float neg_contrib =
      hard_negative_sum(s_con, s_maskw, s_hist, s_scan, sredf, sredi,
                        &s_prefix, &s_rem, k, nz, neg_tot, pos_con, tid);

  if (tid == 0) {
    const float total = loc_loss + pos_con + neg_contrib;
    const float denom = (float)(pos != 0 ? pos : 1);
    rows[b] = total / denom;
  }
}

__global__ void ssd_zero_acc(float* accf, int* acci) {
  const int i = blockIdx.x * blockDim.x + threadIdx.x;
  if (i < 3 * kB) accf[i] = 0.f;
  if (i < 2 * kB) acci[i] = 0;
}

// =================== fallback: monolithic per-row kernel ===================
__global__ __launch_bounds__(NT)
void ssd_loss_rows(const float* __restrict__ loc_out,
                   const float* __restrict__ cls_out,
                   const float* __restrict__ loc_lab,
                   const int*   __restrict__ cls_lab,
                   float*       __restrict__ rows) {
  __shared__ float    s_logits[CH * kC];
  __shared__ float    s_con[kA];
  __shared__ uint32_t s_maskw[NMASKW];
  __shared__ uint32_t s_hist[256];
  __shared__ int      s_scan[NT];
  __shared__ float    sredf[4];
  __shared__ int      sredi[3];
  __shared__ uint32_t s_prefix;
  __shared__ int      s_rem;

  const int b   = blockIdx.x;
  const int tid = threadIdx.x;
  const size_t rowbase = (size_t)b * kA;

  for (int i = tid; i < NMASKW; i += NT) s_maskw[i] = 0u;
  if (tid == 0) { sredf[0] = sredf[1] = sredf[2] = 0.f; sredi[0] = sredi[1] = 0; }
  __syncthreads();

  float acc_loc = 0.f, acc_posc = 0.f, acc_negc = 0.f;
  int   acc_pos = 0,   acc_nz = 0;

  for (int base = 0; base < kA; base += CH) {
    const int cn  = min(CH, kA - base);
    const int nel = cn * kC;
    const float* src = cls_out + (rowbase + base) * (size_t)kC;
#ifdef USE_TDM
    if (tid < 32) {
      tdm_load_1d((uint32_t)(uintptr_t)(&s_logits[0]), src, (uint32_t)nel);
      __builtin_amdgcn_s_wait_tensorcnt(0);
    }
#else
    for (int i = tid; i < nel; i += NT) s_logits[i] = src[i];
#endif
    __syncthreads();

    if (tid < cn) {
      const int a   = base + tid;
      const int lbl = cls_lab[rowbase + a];
      const float ce = cross_entropy_from_lds(&s_logits[tid * kC], lbl);
      s_con[a] = ce;
      if (lbl > 0) {
        atomicOr(&s_maskw[a >> 5], 1u << (a & 31));
        acc_pos  += 1;
        acc_posc += ce;
        acc_loc  += smooth_l1_4(
            reinterpret_cast<const float4*>(loc_out)[rowbase + a],
            reinterpret_cast<const float4*>(loc_lab)[rowbase + a]);
      } else {
        acc_negc += ce;
        if (ce > 0.f) acc_nz += 1;
      }
    }
    __syncthreads();
  }

  atomicAdd(&sredf[0], acc_loc);
  atomicAdd(&sredf[1], acc_posc);
  atomicAdd(&sredf[2], acc_negc);
  atomicAdd(&sredi[0], acc_pos);
  atomicAdd(&sredi[1], acc_nz);
  __syncthreads();
  const float loc_loss = sredf[0];
  const float pos_con  = sredf[1];
  const float neg_tot  = sredf[2];
  const int   pos      = sredi[0];
  const int   nz       = sredi[1];
  __syncthreads();

  int k = min(3 * pos, kA);
  if (k == 0) k = 3;
  const float neg_contrib =
      hard_negative_sum(s_con, s_maskw, s_hist, s_scan, sredf, sredi,
                        &s_prefix, &s_rem, k, nz, neg_tot, pos_con, tid);

  if (tid == 0) {
    const float total = loc_loss + pos_con + neg_contrib;
    const float denom = (float)(pos != 0 ? pos : 1);
    rows[b] = total / denom;
  }
}

__global__ __launch_bounds__(128)
void ssd_loss_mean(const float* __restrict__ rows, float* __restrict__ out) {
  __shared__ float s[128];
  const int t = threadIdx.x;
  s[t] = rows[t];
  __syncthreads();
  for (int d = 64; d > 0; d >>= 1) {
    if (t < d) s[t] += s[t + d];
    __syncthreads();
  }
  if (t == 0) out[0] = s[0] / (float)kB;
}

extern "C" void kernel_launch(void* const* d_in, const int* in_sizes, int n_in,
                              void* d_out, int out_size, void* d_ws,
                              size_t ws_size, hipStream_t stream) {
  (void)in_sizes; (void)n_in; (void)out_size;
  const float* loc_out = (const float*)d_in[0];
  const float* cls_out = (const float*)d_in[1];
  const float* loc_lab = (const float*)d_in[2];
  const int*   cls_lab = (const int*)d_in[3];

  const size_t conElems = (size_t)kB * kA;
  const size_t needBytes = conElems * 4 + (3 * kB) * 4 + (2 * kB) * 4 + kB * 4;

  if (ws_size >= needBytes) {
    // fast path: split phases, full-machine streaming with overlapped TDM
    float* ws_con = (float*)d_ws;
    float* accf   = ws_con + conElems;
    int*   acci   = (int*)(accf + 3 * kB);
    float* rows   = (float*)(acci + 2 * kB);

    ssd_zero_acc<<<2, 256, 0, stream>>>(accf, acci);
    ssd_phaseA<<<dim3(NSEG, kB), NT, 0, stream>>>(loc_out, cls_out, loc_lab,
                                                  cls_lab, ws_con, accf, acci);
    ssd_phaseB<<<kB, NT, 0, stream>>>(ws_con, cls_lab, accf, acci, rows);
    ssd_loss_mean<<<1, 128, 0, stream>>>(rows, (float*)d_out);
  } else {
    // fallback: monolithic per-row kernel (512B workspace)
    float* rows = (float*)d_ws;
    ssd_loss_rows<<<kB, NT, 0, stream>>>(loc_out, cls_out, loc_lab, cls_lab,
                                         rows);
    ssd_loss_mean<<<1, 128, 0, stream>>>(rows, (float*)d_out);
  }
}